// RNN_1047972020632
// MI455X (gfx1250) — compile-verified
//
#include <hip/hip_runtime.h>

#define LSTM_B 256
#define LSTM_T 2000
#define LSTM_I 80
#define LSTM_H 20
#define LSTM_G 80            // 4*H
#define LSTM_BT (LSTM_B * LSTM_T)

typedef float v2f __attribute__((ext_vector_type(2)));
typedef float v8f __attribute__((ext_vector_type(8)));

__device__ __forceinline__ float fast_sig(float x) {
  // 1/(1+exp(-x)) via v_exp_f32 (exp2) + v_rcp_f32
  return __builtin_amdgcn_rcpf(1.0f + __builtin_amdgcn_exp2f(-1.4426950408889634f * x));
}
__device__ __forceinline__ float fast_tanh(float x) {
  // tanh(x) = 2*sigmoid(2x) - 1
  return 2.0f * __builtin_amdgcn_rcpf(1.0f + __builtin_amdgcn_exp2f(-2.8853900817779268f * x)) - 1.0f;
}

// Async-copy one 16x80 f32 slice (5120 B) global -> LDS. Each lane moves 16 B
// per issue, 10 issues sharing ONE LDS base VGPR and ONE global base pair:
// the 24-bit instruction offset is added to both addresses (ISA 7.4.4), so no
// per-issue VALU address math. Tracked by ASYNCcnt. LDS address = low 32 bits
// of the flat shared pointer (ISA 10.2).
#define ASYNC_ISSUE(OFF)                                                    \
  asm volatile("global_load_async_to_lds_b128 %0, %1, off offset:" #OFF     \
               :: "v"(lofs), "v"(g) : "memory")

__device__ __forceinline__ void async_copy_slice(const float* __restrict__ gsrc,
                                                 const float* ldst, int lane) {
  unsigned lofs = (unsigned)(uintptr_t)ldst + (unsigned)lane * 16u;
  const float* g = gsrc + lane * 4;
  ASYNC_ISSUE(0);    ASYNC_ISSUE(512);  ASYNC_ISSUE(1024); ASYNC_ISSUE(1536);
  ASYNC_ISSUE(2048); ASYNC_ISSUE(2560); ASYNC_ISSUE(3072); ASYNC_ISSUE(3584);
  ASYNC_ISSUE(4096); ASYNC_ISSUE(4608);
}

// ---------------------------------------------------------------------------
// XG[d][t][b][g] = sum_k X[b*T+t][k] * W[d][k][g] + Bias[d][g]
// One wave computes two 16-row x 80-col strips (B-fragments shared across the
// two A streams) with fp32 WMMA 16x16x4. Output is time-major so the
// recurrence reads one contiguous 5 KB slice per (tile, step).
// ---------------------------------------------------------------------------
template <int K>
__global__ void __launch_bounds__(128)
xg_gemm_wmma(const float* __restrict__ X, const float* __restrict__ W,
             const float* __restrict__ Bias, float* __restrict__ XGT) {
  const int lane = threadIdx.x & 31;
  const int wid  = blockIdx.x * (blockDim.x >> 5) + (threadIdx.x >> 5);
  const int d    = blockIdx.y;
  const int m0   = wid * 32;            // two 16-row tiles per wave
  const int colb = lane & 15;
  const int koff = (lane >> 4) << 1;    // A/B frag K sub-offset: 0 or 2
  const int mh   = (lane >> 4) << 3;    // C/D frag M half: 0 or 8

  const float* __restrict__ Wd = W + (size_t)d * K * LSTM_G;

  v8f acc[2][5];
  const v8f vzero = {0.f, 0.f, 0.f, 0.f, 0.f, 0.f, 0.f, 0.f};
#pragma unroll
  for (int mt = 0; mt < 2; ++mt)
#pragma unroll
    for (int n = 0; n < 5; ++n) acc[mt][n] = vzero;

  const float* __restrict__ xrow0 = X + (size_t)(m0 + colb) * K;
  const float* __restrict__ xrow1 = X + (size_t)(m0 + 16 + colb) * K;

  for (int k0 = 0; k0 < K; k0 += 4) {
    v2f bf[5];
#pragma unroll
    for (int n = 0; n < 5; ++n) {
      const float* wp = Wd + (size_t)(k0 + koff) * LSTM_G + n * 16 + colb;
      bf[n].x = wp[0];
      bf[n].y = wp[LSTM_G];
    }
    v2f a0, a1;
    a0.x = xrow0[k0 + koff];
    a0.y = xrow0[k0 + koff + 1];
    a1.x = xrow1[k0 + koff];
    a1.y = xrow1[k0 + koff + 1];
#pragma unroll
    for (int n = 0; n < 5; ++n)
      acc[0][n] = __builtin_amdgcn_wmma_f32_16x16x4_f32(
          false, a0, false, bf[n], (short)0, acc[0][n], false, false);
#pragma unroll
    for (int n = 0; n < 5; ++n)
      acc[1][n] = __builtin_amdgcn_wmma_f32_16x16x4_f32(
          false, a1, false, bf[n], (short)0, acc[1][n], false, false);
  }

#pragma unroll
  for (int mt = 0; mt < 2; ++mt) {
    const int mbase = m0 + mt * 16;
    const int b  = mbase / LSTM_T;   // T % 16 == 0: one batch index per tile
    const int t0 = mbase % LSTM_T;
#pragma unroll
    for (int n = 0; n < 5; ++n) {
      const int col = n * 16 + colb;
      const float bv = Bias[d * LSTM_G + col];
#pragma unroll
      for (int r = 0; r < 8; ++r) {
        const int t = t0 + mh + r;
        XGT[(((size_t)d * LSTM_T + t) * LSTM_B + b) * LSTM_G + col] =
            acc[mt][n][r] + bv;
      }
    }
  }
}

// ---------------------------------------------------------------------------
// Sequential scan. One wave owns a 16-batch tile for one direction.
// - xg slices are async-copied (ASYNCcnt) straight into a double-buffered
//   LDS gates buffer; the wait is deferred until after the WMMA chain so
//   async latency hides behind compute.
// - WMMA chain h @ Whh starts from the inline-0 C operand (no C assembly),
//   then each lane ds_add_f32's its D fragment into the xg-filled buffer:
//   gates = xg + h @ Whh entirely in LDS, no register shuffling.
// - h tile in LDS (feeds A fragments), c in registers, Whh frags in registers.
// ---------------------------------------------------------------------------
__global__ void __launch_bounds__(32)
lstm_recur_wmma(const float* __restrict__ XGT, const float* __restrict__ Whh,
                float* __restrict__ Hout) {
  __shared__ float gates_lds[2][16 * LSTM_G];   // xg + h@Whh, double-buffered
  __shared__ float h_lds[16 * LSTM_H];

  const int lane = threadIdx.x;
  const int b0   = blockIdx.x * 16;
  const int d    = blockIdx.y;
  const int colb = lane & 15;
  const int koff = (lane >> 4) << 1;
  const int mh   = (lane >> 4) << 3;
  const int m    = lane & 15;        // batch row owned for activations
  const int j0   = (lane >> 4) * 10; // hidden-unit range owned

  // Preload all 25 Whh B-fragments (K=20 -> 5 k-tiles, G=80 -> 5 n-tiles)
  v2f bw[5][5];
  const float* __restrict__ Wd = Whh + (size_t)d * LSTM_H * LSTM_G;
#pragma unroll
  for (int kt = 0; kt < 5; ++kt)
#pragma unroll
    for (int n = 0; n < 5; ++n) {
      const float* wp = Wd + (size_t)(kt * 4 + koff) * LSTM_G + n * 16 + colb;
      bw[kt][n].x = wp[0];
      bw[kt][n].y = wp[LSTM_G];
    }

  // Zero initial h and c state
#pragma unroll
  for (int q = 0; q < 10; ++q) h_lds[lane * 10 + q] = 0.0f;
  float c_reg[10];
#pragma unroll
  for (int q = 0; q < 10; ++q) c_reg[q] = 0.0f;
  __syncthreads();

  // Kick off async copy of step 0's slice into buffer 0
  {
    const int t_first = (d == 0) ? 0 : (LSTM_T - 1);
    async_copy_slice(
        XGT + (((size_t)d * LSTM_T + t_first) * LSTM_B + b0) * LSTM_G,
        &gates_lds[0][0], lane);
  }

  const v8f vzero = {0.f, 0.f, 0.f, 0.f, 0.f, 0.f, 0.f, 0.f};

  for (int step = 0; step < LSTM_T; ++step) {
    const int tt  = (d == 0) ? step : (LSTM_T - 1 - step);
    const int buf = step & 1;

    // Issue next slice into the other buffer (wait is deferred past WMMA)
    const bool have_next = (step + 1 < LSTM_T);
    if (have_next) {
      const int tn = (d == 0) ? (step + 1) : (LSTM_T - 2 - step);
      async_copy_slice(
          XGT + (((size_t)d * LSTM_T + tn) * LSTM_B + b0) * LSTM_G,
          &gates_lds[buf ^ 1][0], lane);
    }

    // A fragments for this step's h (depends only on LDS h, not async data)
    v2f af[5];
#pragma unroll
    for (int kt = 0; kt < 5; ++kt) {
      af[kt].x = h_lds[m * LSTM_H + kt * 4 + koff];
      af[kt].y = h_lds[m * LSTM_H + kt * 4 + koff + 1];
    }

    // h @ Whh with C = inline 0, then chained accumulation
    v8f g[5];
#pragma unroll
    for (int n = 0; n < 5; ++n)
      g[n] = __builtin_amdgcn_wmma_f32_16x16x4_f32(
          false, af[0], false, bw[0][n], (short)0, vzero, false, false);
#pragma unroll
    for (int kt = 1; kt < 5; ++kt)
#pragma unroll
      for (int n = 0; n < 5; ++n)
        g[n] = __builtin_amdgcn_wmma_f32_16x16x4_f32(
            false, af[kt], false, bw[kt][n], (short)0, g[n], false, false);

    __syncthreads();  // h_lds reads (above) vs h_lds writes (below)

    // Current slice must have landed before we accumulate into it; the 10
    // just-issued loads for the next slice may remain outstanding.
    if (have_next) {
      asm volatile("s_wait_asynccnt 10" ::: "memory");
    } else {
      asm volatile("s_wait_asynccnt 0" ::: "memory");
    }

    // gates_lds[buf] already holds xg(t); add the D fragments in place
    float* __restrict__ gb = &gates_lds[buf][0];
#pragma unroll
    for (int n = 0; n < 5; ++n) {
      const int col = n * 16 + colb;
#pragma unroll
      for (int r = 0; r < 8; ++r)
        atomicAdd(&gb[(mh + r) * LSTM_G + col], g[n][r]);  // ds_add_f32
    }
    __syncthreads();

    // Activations: gate order i,f,g,o (widths 20); each lane owns 10 cells
    float hn_arr[10];
#pragma unroll
    for (int q = 0; q < 10; ++q) {
      const int j = j0 + q;
      const float iv = gb[m * LSTM_G + j];
      const float fv = gb[m * LSTM_G + 20 + j];
      const float gv = gb[m * LSTM_G + 40 + j];
      const float ov = gb[m * LSTM_G + 60 + j];
      const float cn = fast_sig(fv) * c_reg[q] + fast_sig(iv) * fast_tanh(gv);
      const float hn = fast_sig(ov) * fast_tanh(cn);
      c_reg[q] = cn;
      hn_arr[q] = hn;
      h_lds[m * LSTM_H + j] = hn;
    }
    // Layer output, batch-major (B, T, 2H): lane's 10 floats are contiguous
    // and 8B-aligned -> 5 x b64 stores
    {
      float2* __restrict__ hp = (float2*)(
          Hout + ((size_t)(b0 + m) * LSTM_T + tt) * (2 * LSTM_H) +
          d * LSTM_H + j0);
#pragma unroll
      for (int p = 0; p < 5; ++p)
        hp[p] = make_float2(hn_arr[2 * p], hn_arr[2 * p + 1]);
    }
    __syncthreads();
  }
}

// ---------------------------------------------------------------------------
// out[b*T+t] = H1[b,t,:] . fc_w + fc_b
// ---------------------------------------------------------------------------
__global__ void __launch_bounds__(256)
fc_kernel(const float* __restrict__ Hin, const float* __restrict__ fcw,
          const float* __restrict__ fcb, float* __restrict__ out) {
  const int idx = blockIdx.x * blockDim.x + threadIdx.x;
  if (idx >= LSTM_BT) return;
  const float* h = Hin + (size_t)idx * (2 * LSTM_H);
  float acc = fcb[0];
#pragma unroll
  for (int j = 0; j < 2 * LSTM_H; ++j) acc += h[j] * fcw[j];
  out[idx] = acc;
}

extern "C" void kernel_launch(void* const* d_in, const int* in_sizes, int n_in,
                              void* d_out, int out_size, void* d_ws, size_t ws_size,
                              hipStream_t stream) {
  (void)in_sizes; (void)n_in; (void)out_size; (void)ws_size;
  const float* x    = (const float*)d_in[0];
  const float* wih0 = (const float*)d_in[1];
  const float* whh0 = (const float*)d_in[2];
  const float* bb0  = (const float*)d_in[3];
  const float* wih1 = (const float*)d_in[4];
  const float* whh1 = (const float*)d_in[5];
  const float* bb1  = (const float*)d_in[6];
  const float* fcw  = (const float*)d_in[7];
  const float* fcb  = (const float*)d_in[8];
  float* out = (float*)d_out;

  // Workspace: XGT (2,T,B,80) = 327.7 MB ; Hbuf (B,T,40) = 81.9 MB (reused)
  float* XGT  = (float*)d_ws;
  float* Hbuf = XGT + (size_t)2 * LSTM_T * LSTM_B * LSTM_G;

  const dim3 gblk(128);                          // 4 waves / block
  const dim3 ggrd(LSTM_BT / 32 / 4, 2);          // two 16-row strips per wave
  const dim3 rgrd(LSTM_B / 16, 2);               // 16-batch tile per wave

  // Layer 0
  xg_gemm_wmma<LSTM_I><<<ggrd, gblk, 0, stream>>>(x, wih0, bb0, XGT);
  lstm_recur_wmma<<<rgrd, 32, 0, stream>>>(XGT, whh0, Hbuf);
  // Layer 1 (input = layer-0 hidden, K = 40)
  xg_gemm_wmma<2 * LSTM_H><<<ggrd, gblk, 0, stream>>>(Hbuf, wih1, bb1, XGT);
  lstm_recur_wmma<<<rgrd, 32, 0, stream>>>(XGT, whh1, Hbuf);
  // Final projection
  fc_kernel<<<(LSTM_BT + 255) / 256, 256, 0, stream>>>(Hbuf, fcw, fcb, out);
}